// Circular_Pooling_36507222016327
// MI455X (gfx1250) — compile-verified
//
#include <hip/hip_runtime.h>

#ifndef __has_builtin
#define __has_builtin(x) 0
#endif

#if defined(__HIP_DEVICE_COMPILE__) && \
    __has_builtin(__builtin_amdgcn_global_load_async_to_lds_b128) && \
    __has_builtin(__builtin_amdgcn_s_wait_asynccnt)
#define CP_USE_ASYNC 1
typedef int cp_v4i __attribute__((vector_size(4 * sizeof(int))));
typedef __attribute__((address_space(1))) cp_v4i* cp_gptr;   // global (AS1) int4*
typedef __attribute__((address_space(3))) cp_v4i* cp_lptr;   // LDS (AS3) int4*
#else
#define CP_USE_ASYNC 0
#endif

// Problem dimensions (fixed by the reference)
#define DIM_B      64
#define DIM_C      64
#define DIM_H      15
#define DIM_W      15
#define DIM_K      200
#define DIM_OH     13
#define DIM_OW     13
#define DIM_K2     100

#define KC         40                  // input channels staged per block (160B -> 16B aligned)
#define KC2        (KC / 2)            // 20 output channels per block
#define NCHUNK     (DIM_K / KC)        // 5
#define PLANE_IN   (DIM_H * DIM_W * DIM_K)    // 45000 floats
#define PLANE_OUT  (DIM_OH * DIM_OW * DIM_K2) // 16900 floats
#define NPOS       (DIM_H * DIM_W)            // 225 spatial positions
#define N4         (NPOS * (KC / 4))          // 2250 float4 copies per block
#define NOUT       (DIM_OH * DIM_OW * KC2)    // 3380 outputs per block
#define NTHREADS   256

__global__ __launch_bounds__(NTHREADS)
void circular_pool_kernel(const float* __restrict__ x, float* __restrict__ out)
{
    // 15 x 15 positions x 40 channels = 9000 floats = 36 KB LDS
    __shared__ float lds[NPOS * KC];

    const int bid   = blockIdx.x;
    const int plane = bid / NCHUNK;          // (b*64 + c), 0..4095
    const int chunk = bid - plane * NCHUNK;  // 0..4
    const int tid   = threadIdx.x;

    // ---------------- Stage: global -> LDS (async copy path, CDNA5) ----------------
    const float* gbase = x + (size_t)plane * PLANE_IN + chunk * KC;

    for (int idx = tid; idx < N4; idx += NTHREADS) {
        const int p = idx / (KC / 4);        // spatial position 0..224
        const int q = idx - p * (KC / 4);    // float4 index within chunk, 0..9
        const float* g = gbase + p * DIM_K + q * 4;   // 16B-aligned
        float*       l = lds   + p * KC    + q * 4;   // 16B-aligned
#if CP_USE_ASYNC
        // one instruction per wave moves 16B per active lane, tracked by ASYNCcnt
        __builtin_amdgcn_global_load_async_to_lds_b128((cp_gptr)g, (cp_lptr)l, 0, 0);
#else
        *(float4*)l = *(const float4*)g;
#endif
    }
#if CP_USE_ASYNC
    __builtin_amdgcn_s_wait_asynccnt(0);
#endif
    __syncthreads();

    // ---------------- Compute: 13x13x20 outputs from LDS ----------------
    float* obase = out + (size_t)plane * PLANE_OUT + chunk * KC2;

    for (int o = tid; o < NOUT; o += NTHREADS) {
        const int k2 = o % KC2;              // output channel within chunk, 0..19
        const int t  = o / KC2;              // spatial index 0..168
        const int j  = t % DIM_OW;
        const int i  = t / DIM_OW;

        // circular-pool window corners; duplicates when the window is size 1
        // make the divisor uniformly 8 (see analysis).
        const int ra = (i < 6) ? i     : ((i == 6) ? 7 : i + 1);
        const int rb = (i < 6) ? i + 1 : ((i == 6) ? 7 : i + 2);
        const int ca = (j < 6) ? j     : ((j == 6) ? 7 : j + 1);
        const int cb = (j < 6) ? j + 1 : ((j == 6) ? 7 : j + 2);
        const int kk = 2 * k2;               // even -> 8B-aligned float2 in LDS

        const float2 s00 = *(const float2*)&lds[(ra * DIM_W + ca) * KC + kk];
        const float2 s01 = *(const float2*)&lds[(ra * DIM_W + cb) * KC + kk];
        const float2 s10 = *(const float2*)&lds[(rb * DIM_W + ca) * KC + kk];
        const float2 s11 = *(const float2*)&lds[(rb * DIM_W + cb) * KC + kk];

        const float sx = (s00.x + s01.x) + (s10.x + s11.x);
        const float sy = (s00.y + s01.y) + (s10.y + s11.y);

        obase[(i * DIM_OW + j) * DIM_K2 + k2] = (sx + sy) * 0.125f;
    }
}

extern "C" void kernel_launch(void* const* d_in, const int* in_sizes, int n_in,
                              void* d_out, int out_size, void* d_ws, size_t ws_size,
                              hipStream_t stream) {
    (void)in_sizes; (void)n_in; (void)d_ws; (void)ws_size; (void)out_size;
    const float* x = (const float*)d_in[0];
    float* out = (float*)d_out;

    const int nblocks = DIM_B * DIM_C * NCHUNK;  // 4096 * 5 = 20480
    circular_pool_kernel<<<nblocks, NTHREADS, 0, stream>>>(x, out);
}